// SimpleGNN_163208757904
// MI455X (gfx1250) — compile-verified
//
#include <hip/hip_runtime.h>

// ---------------------------------------------------------------------------
// SimpleGNN for MI455X (gfx1250): WMMA f32 16x16x4 GEMMs + atomic GCN scatter
// ---------------------------------------------------------------------------

typedef __attribute__((ext_vector_type(2))) float v2f;
typedef __attribute__((ext_vector_type(8))) float v8f;

__device__ __forceinline__ float gelu_exact(float x) {
  // jax.nn.gelu default (exact): 0.5*x*(1+erf(x/sqrt(2)))
  return 0.5f * x * (1.0f + erff(x * 0.70710678118654752440f));
}

// Y[n,16] = (FUSE? gelu(X@W + bias) : X@W), X:[n,KIN], W:[KIN,16] row-major.
// One wave (32 lanes) computes one 16-row tile via V_WMMA_F32_16X16X4_F32.
// ISA 32-bit A layout (16x4): lanes 0-15 hold row M=lane with K={0,1} in the
// two A VGPRs; lanes 16-31 hold the same rows with K={2,3}. B (4x16) mirrors
// this with N striped across lanes. C/D: VGPR r -> D[M=8*hi+r][N=col].
template <int KIN, bool FUSE>
__global__ void k_gemm16(const float* __restrict__ X,
                         const float* __restrict__ W,
                         const float* __restrict__ bias,
                         float* __restrict__ Y, int n) {
  int wid  = (blockIdx.x * blockDim.x + threadIdx.x) >> 5;
  int lane = threadIdx.x & 31;
  int row0 = wid << 4;
  if (row0 >= n) return;          // wave-uniform exit: EXEC all-1s for WMMA
  int col = lane & 15;
  int hi  = lane >> 4;            // 0: K pair {0,1}; 1: K pair {2,3} of slice
  int arow = row0 + col;          // A-matrix row held by this lane

  v8f c = {0.f, 0.f, 0.f, 0.f, 0.f, 0.f, 0.f, 0.f};

  if constexpr (KIN == 16) {
    // whole A row per lane: 4 x global_load_b128, fragments via lane-half select
    const float4* xr = (const float4*)(X + (size_t)arow * 16);
    float4 q0 = xr[0], q1 = xr[1], q2 = xr[2], q3 = xr[3];
#pragma unroll
    for (int j = 0; j < 4; ++j) {
      float4 q = (j == 0) ? q0 : (j == 1) ? q1 : (j == 2) ? q2 : q3;
      v2f a, b;
      a.x = hi ? q.z : q.x;
      a.y = hi ? q.w : q.y;
      int k0 = (j << 2) + (hi << 1);
      b.x = W[k0 * 16 + col];
      b.y = W[k0 * 16 + 16 + col];
      c = __builtin_amdgcn_wmma_f32_16x16x4_f32(false, a, false, b,
                                                (short)0, c, false, false);
    }
  } else {  // KIN == 2: single WMMA, upper K half zero-padded (no divergence)
    float2 q = *(const float2*)(X + (size_t)arow * 2);
    v2f a, b;
    a.x = hi ? 0.f : q.x;
    a.y = hi ? 0.f : q.y;
    b.x = hi ? 0.f : W[col];
    b.y = hi ? 0.f : W[16 + col];
    c = __builtin_amdgcn_wmma_f32_16x16x4_f32(false, a, false, b,
                                              (short)0, c, false, false);
  }

#pragma unroll
  for (int r = 0; r < 8; ++r) {
    int row = row0 + (hi << 3) + r;
    float v = c[r];
    if constexpr (FUSE) v = gelu_exact(v + bias[col]);
    Y[row * 16 + col] = v;
  }
}

// ----------------------------- GCN helpers ---------------------------------

__global__ void k_init_deg(float* deg, int n) {
  int i = blockIdx.x * blockDim.x + threadIdx.x;
  if (i < n) deg[i] = 1.0f;   // self-loop weight
}

__global__ void k_edge_deg(const int* __restrict__ src, const int* __restrict__ dst,
                           const float* __restrict__ w, const int* __restrict__ mapping,
                           float* deg, int ne) {
  int e = blockIdx.x * blockDim.x + threadIdx.x;
  if (e >= ne) return;
  int d = dst[e];
  if (mapping) {
    int s = mapping[src[e]];
    d = mapping[d];
    if (s < 0 || d < 0) return;   // edge dropped by pooling (weight -> 0)
  }
  atomicAdd(&deg[d], w[e]);
}

__global__ void k_dinv(float* deg, int n) {   // in-place deg -> deg^-1/2
  int i = blockIdx.x * blockDim.x + threadIdx.x;
  if (i < n) {
    float d = deg[i];
    deg[i] = (d > 0.0f) ? rsqrtf(d) : 0.0f;
  }
}

__global__ void k_zero(float* p, int n) {
  int i = blockIdx.x * blockDim.x + threadIdx.x;
  if (i < n) p[i] = 0.0f;
}

__global__ void k_edge_scatter(const int* __restrict__ src, const int* __restrict__ dst,
                               const float* __restrict__ w, const int* __restrict__ mapping,
                               const float* __restrict__ dinv,
                               const float* __restrict__ hw,
                               float* __restrict__ out, int ne) {
  int e = blockIdx.x * blockDim.x + threadIdx.x;
  if (e >= ne) return;
  int s = src[e], d = dst[e];
  if (mapping) {
    s = mapping[s]; d = mapping[d];
    if (s < 0 || d < 0) return;
  }
  float norm = dinv[s] * w[e] * dinv[d];
  if (norm == 0.0f) return;
  const float4* hs = (const float4*)(hw + (size_t)s * 16);
  float* od = out + (size_t)d * 16;
#pragma unroll
  for (int q = 0; q < 4; ++q) {
    float4 v = hs[q];
    atomicAdd(od + q * 4 + 0, v.x * norm);
    atomicAdd(od + q * 4 + 1, v.y * norm);
    atomicAdd(od + q * 4 + 2, v.z * norm);
    atomicAdd(od + q * 4 + 3, v.w * norm);
  }
}

// Y = gelu(Y + hw*dinv^2 + bias) (+ res); Y holds the edge-aggregated sums.
__global__ void k_finalize(float* __restrict__ Y, const float* __restrict__ hw,
                           const float* __restrict__ dinv,
                           const float* __restrict__ bias,
                           const float* __restrict__ res, int n) {
  int t = blockIdx.x * blockDim.x + threadIdx.x;
  if (t >= n * 16) return;
  int i = t >> 4, f = t & 15;
  float di = dinv[i];
  float v = Y[t] + hw[t] * di * di + bias[f];
  v = gelu_exact(v);
  if (res) v += res[t];
  Y[t] = v;
}

// ---------------------------- TopK pooling ---------------------------------
// scal layout (one 256B ws slot): f[0]=lo, f[1]=hi, f[2]=mid, f[3]=1/||p||,
// ((int*)scal)[4] = counter.

__global__ void k_prep(const float* __restrict__ p, float* scal) {
  if (blockIdx.x == 0 && threadIdx.x == 0) {
    float s = 0.0f;
    for (int f = 0; f < 16; ++f) s += p[f] * p[f];
    scal[3] = rsqrtf(s);
    scal[0] = -2.0f;   // count(score > lo) = N >= K
    scal[1] =  2.0f;   // count(score > hi) = 0 <= K
  }
}

__global__ void k_score(const float* __restrict__ h, const float* __restrict__ p,
                        const float* __restrict__ scal, float* score, int n) {
  int i = blockIdx.x * blockDim.x + threadIdx.x;
  if (i >= n) return;
  float s = 0.0f;
#pragma unroll
  for (int f = 0; f < 16; ++f) s += h[i * 16 + f] * p[f];
  score[i] = tanhf(s * scal[3]);
}

__global__ void k_bisect_mid(float* scal) {
  if (blockIdx.x == 0 && threadIdx.x == 0) {
    scal[2] = 0.5f * (scal[0] + scal[1]);
    ((int*)scal)[4] = 0;
  }
}

__global__ void k_count_gt(const float* __restrict__ score, float* scal, int n) {
  int i = blockIdx.x * blockDim.x + threadIdx.x;
  if (i >= n) return;
  if (score[i] > scal[2]) atomicAdd(((int*)scal) + 4, 1);
}

__global__ void k_bisect_update(float* scal, int k) {
  if (blockIdx.x == 0 && threadIdx.x == 0) {
    int c = ((int*)scal)[4];
    if (c >= k) scal[0] = scal[2];   // keep count(>lo) >= k
    else        scal[1] = scal[2];   // keep count(>hi) <= k
  }
}

// Single-block (1024 threads) deterministic compaction:
// pass 0: score > hi (strictly above threshold), in index order;
// pass 1: lo < score <= hi boundary/tie fill up to k.
__global__ void k_compact(const float* __restrict__ score, int n, int k,
                          const float* __restrict__ scal, int* __restrict__ perm) {
  __shared__ int sbuf[1024];
  __shared__ int sbase;
  float lo = scal[0], hi = scal[1];
  int tid = threadIdx.x;
  if (tid == 0) sbase = 0;
  __syncthreads();
  for (int pass = 0; pass < 2; ++pass) {
    for (int start = 0; start < n; start += 1024) {
      int i = start + tid;
      int flag = 0;
      if (i < n) {
        float s = score[i];
        flag = (pass == 0) ? (s > hi) : (s <= hi && s > lo);
      }
      sbuf[tid] = flag;
      __syncthreads();
      for (int off = 1; off < 1024; off <<= 1) {
        int v = (tid >= off) ? sbuf[tid - off] : 0;
        __syncthreads();
        sbuf[tid] += v;
        __syncthreads();
      }
      int pos = sbase + sbuf[tid] - 1;
      if (flag && pos < k) perm[pos] = i;
      __syncthreads();
      if (tid == 0) sbase += sbuf[1023];
      __syncthreads();
    }
  }
}

__global__ void k_map_init(int* mapping, int n) {
  int i = blockIdx.x * blockDim.x + threadIdx.x;
  if (i < n) mapping[i] = -1;
}

__global__ void k_map_set(const int* __restrict__ perm, int* mapping, int k) {
  int j = blockIdx.x * blockDim.x + threadIdx.x;
  if (j < k) mapping[perm[j]] = j;
}

__global__ void k_pool_apply(const float* __restrict__ h, const float* __restrict__ score,
                             const int* __restrict__ perm, float* __restrict__ xo, int k) {
  int t = blockIdx.x * blockDim.x + threadIdx.x;
  if (t >= k * 16) return;
  int j = t >> 4, f = t & 15;
  int i = perm[j];
  xo[t] = h[i * 16 + f] * score[i];
}

__global__ void k_down(const float* __restrict__ h, const float* __restrict__ wd,
                       const float* __restrict__ bd, float* __restrict__ out, int k) {
  int j = blockIdx.x * blockDim.x + threadIdx.x;
  if (j >= k) return;
  float s = bd[0];
#pragma unroll
  for (int f = 0; f < 16; ++f) s += h[j * 16 + f] * wd[f];
  out[j] = s;
}

// ---------------------------------------------------------------------------

extern "C" void kernel_launch(void* const* d_in, const int* in_sizes, int n_in,
                              void* d_out, int out_size, void* d_ws, size_t ws_size,
                              hipStream_t stream) {
  (void)n_in; (void)out_size; (void)ws_size;
  const float* x      = (const float*)d_in[0];
  const int*   ei     = (const int*)d_in[1];
  const float* ea     = (const float*)d_in[2];
  const float* w_up1  = (const float*)d_in[3];
  const float* b_up1  = (const float*)d_in[4];
  const float* w_up2  = (const float*)d_in[5];
  const float* b_up2  = (const float*)d_in[6];
  const float* w_g[5] = {(const float*)d_in[7],  (const float*)d_in[9],
                         (const float*)d_in[11], (const float*)d_in[13],
                         (const float*)d_in[15]};
  const float* b_g[5] = {(const float*)d_in[8],  (const float*)d_in[10],
                         (const float*)d_in[12], (const float*)d_in[14],
                         (const float*)d_in[16]};
  const float* p_pool = (const float*)d_in[17];
  const float* w_down = (const float*)d_in[18];
  const float* b_down = (const float*)d_in[19];

  const int N = in_sizes[0] / 2;
  const int E = in_sizes[1] / 2;
  const int K = N / 2;
  const int* src = ei;
  const int* dst = ei + E;

  // workspace carve-up (~41 MB)
  char* base = (char*)d_ws;
  auto carve = [&](size_t bytes) -> void* {
    void* p = (void*)base;
    base += (bytes + 255) & ~(size_t)255;
    return p;
  };
  float* B0      = (float*)carve((size_t)N * 16 * sizeof(float));
  float* B1      = (float*)carve((size_t)N * 16 * sizeof(float));
  float* B2      = (float*)carve((size_t)N * 16 * sizeof(float));
  float* deg     = (float*)carve((size_t)N * sizeof(float));
  float* score   = (float*)carve((size_t)N * sizeof(float));
  int*   perm    = (int*)  carve((size_t)K * sizeof(int));
  int*   mapping = (int*)  carve((size_t)N * sizeof(int));
  float* scal    = (float*)carve(256);

  const int BT = 256;
  auto g1 = [&](long long t) { return (int)((t + BT - 1) / BT); };
  auto gemmGrid = [&](int n) { return (int)((((long long)(n + 15) / 16) * 32 + BT - 1) / BT); };

  auto conv = [&](const float* xin, const float* Wg, const float* bg,
                  const int* map, int nn, float* hwbuf, float* outbuf,
                  const float* res) {
    k_gemm16<16, false><<<gemmGrid(nn), BT, 0, stream>>>(xin, Wg, nullptr, hwbuf, nn);
    k_init_deg<<<g1(nn), BT, 0, stream>>>(deg, nn);
    k_edge_deg<<<g1(E), BT, 0, stream>>>(src, dst, ea, map, deg, E);
    k_dinv<<<g1(nn), BT, 0, stream>>>(deg, nn);
    k_zero<<<g1((long long)nn * 16), BT, 0, stream>>>(outbuf, nn * 16);
    k_edge_scatter<<<g1(E), BT, 0, stream>>>(src, dst, ea, map, deg, hwbuf, outbuf, E);
    k_finalize<<<g1((long long)nn * 16), BT, 0, stream>>>(outbuf, hwbuf, deg, bg, res, nn);
  };

  // MLP up-projection (fused bias+gelu in the WMMA epilogue)
  k_gemm16<2,  true><<<gemmGrid(N), BT, 0, stream>>>(x,  w_up1, b_up1, B0, N);
  k_gemm16<16, true><<<gemmGrid(N), BT, 0, stream>>>(B0, w_up2, b_up2, B1, N);

  // conv1 on full graph, residual: h3 = gelu(gcn(h2)) + h2  -> B2
  conv(B1, w_g[0], b_g[0], nullptr, N, B0, B2, B1);

  // TopK pooling (deterministic device-side threshold bisection + scan compact)
  k_prep<<<1, 1, 0, stream>>>(p_pool, scal);
  k_score<<<g1(N), BT, 0, stream>>>(B2, p_pool, scal, score, N);
  for (int it = 0; it < 40; ++it) {
    k_bisect_mid<<<1, 1, 0, stream>>>(scal);
    k_count_gt<<<g1(N), BT, 0, stream>>>(score, scal, N);
    k_bisect_update<<<1, 1, 0, stream>>>(scal, K);
  }
  k_compact<<<1, 1024, 0, stream>>>(score, N, K, scal, perm);
  k_map_init<<<g1(N), BT, 0, stream>>>(mapping, N);
  k_map_set<<<g1(K), BT, 0, stream>>>(perm, mapping, K);
  k_pool_apply<<<g1((long long)K * 16), BT, 0, stream>>>(B2, score, perm, B0, K);

  // pooled convs (edges remapped on the fly through `mapping`)
  conv(B0, w_g[1], b_g[1], mapping, K, B1, B2, nullptr); // h4 -> B2
  conv(B2, w_g[2], b_g[2], mapping, K, B0, B1, B2);      // h5 -> B1
  conv(B1, w_g[3], b_g[3], mapping, K, B2, B0, B1);      // h6 -> B0
  conv(B0, w_g[4], b_g[4], mapping, K, B1, B2, B0);      // h7 -> B2

  k_down<<<g1(K), BT, 0, stream>>>(B2, w_down, b_down, (float*)d_out, K);
}